// MoSARA_39058432590135
// MI455X (gfx1250) — compile-verified
//
#include <hip/hip_runtime.h>
#include <math.h>

typedef float v2f __attribute__((ext_vector_type(2)));
typedef float v4f __attribute__((ext_vector_type(4)));
typedef float v8f __attribute__((ext_vector_type(8)));

// GCC-style vector to match the async builtin's prototype exactly.
typedef int v4i_ __attribute__((vector_size(16)));
typedef __attribute__((address_space(1))) v4i_ g_v4i;   // global (AS1)
typedef __attribute__((address_space(3))) v4i_ l_v4i;   // LDS (AS3)

#define N_ 16384
#define D_ 2048
#define K_ 128
#define E_ 8

#define LDA 36    // padded LDS row stride (floats): 144B = 16B aligned, conflict-free
#define LDB 132   // contract-major B stride for k1 (528B, 16B aligned)

// CDNA5 async global->LDS path (ASYNCcnt). Guarded so the file always compiles;
// fallback is the synchronous VGPR round-trip.
#if defined(__gfx1250__) &&                                                    \
    __has_builtin(__builtin_amdgcn_global_load_async_to_lds_b128) &&           \
    __has_builtin(__builtin_amdgcn_s_wait_asynccnt)
#define USE_ASYNC 1
#else
#define USE_ASYNC 0
#endif

__device__ __forceinline__ void cp_async16(const float* g, float* l)
{
#if USE_ASYNC
    __builtin_amdgcn_global_load_async_to_lds_b128((g_v4i*)g, (l_v4i*)l, 0, 0);
#else
    *(v4f*)l = *(const v4f*)g;
#endif
}

// pending==1: the 8 async ops of the *next* chunk may remain outstanding.
__device__ __forceinline__ void cp_wait(int pending)
{
#if USE_ASYNC
    if (pending) __builtin_amdgcn_s_wait_asynccnt(8);
    else         __builtin_amdgcn_s_wait_asynccnt(0);
#else
    (void)pending;
#endif
}

// ---------------------------------------------------------------------------
// Kernel 1: dst[N,128] = x[N,D] @ B[D,128]   (B = U_k or V_k, contract-major)
// Block tile 128(M) x 128(N), KC=32, double-buffered LDS + async copies.
// 256 threads = 8 waves (4x2), each wave: 32x64 = 2x4 WMMA 16x16 tiles.
// ---------------------------------------------------------------------------
__global__ __launch_bounds__(256, 2)
void xu_gemm_kernel(const float* __restrict__ A,
                    const float* __restrict__ Bm,
                    float* __restrict__ dst)
{
    extern __shared__ float smem[];
    float* AsBase = smem;                    // 2 x [128*LDA]
    float* BsBase = smem + 2 * 128 * LDA;    // 2 x [32*LDB]

    const int tid  = threadIdx.x;
    const int lane = tid & 31;
    const int wid  = tid >> 5;
    const int kh   = lane >> 4;       // half-wave selects K pair
    const int lm   = lane & 15;       // M (A) or N (B/C) index
    const int wm   = (wid >> 1) * 32;
    const int wn   = (wid & 1) * 64;
    const int m0   = blockIdx.x * 128;

    v8f c[2][4];
    #pragma unroll
    for (int im = 0; im < 2; ++im)
        #pragma unroll
        for (int in = 0; in < 4; ++in)
            #pragma unroll
            for (int q = 0; q < 8; ++q) c[im][in][q] = 0.0f;

    auto issue = [&](int ch, int buf) {
        const int c0 = ch * 32;
        float* As = AsBase + buf * (128 * LDA);
        float* Bs = BsBase + buf * (32 * LDB);
        #pragma unroll
        for (int i = 0; i < 4; ++i) {
            int idx = tid + i * 256;
            int r = idx >> 3, c4 = (idx & 7) * 4;
            cp_async16(A + (size_t)(m0 + r) * D_ + c0 + c4, &As[r * LDA + c4]);
        }
        #pragma unroll
        for (int i = 0; i < 4; ++i) {
            int idx = tid + i * 256;
            int r = idx >> 5, c4 = (idx & 31) * 4;
            cp_async16(Bm + (size_t)(c0 + r) * K_ + c4, &Bs[r * LDB + c4]);
        }
    };

    const int CH = D_ / 32;
    issue(0, 0);
    for (int ch = 0; ch < CH; ++ch) {
        if (ch + 1 < CH) { issue(ch + 1, (ch + 1) & 1); cp_wait(1); }
        else             { cp_wait(0); }
        __syncthreads();   // chunk ch resident in LDS for all waves

        const float* As = AsBase + (ch & 1) * (128 * LDA);
        const float* Bs = BsBase + (ch & 1) * (32 * LDB);
        #pragma unroll
        for (int kk = 0; kk < 32; kk += 4) {
            v2f a[2], b[4];
            #pragma unroll
            for (int im = 0; im < 2; ++im)   // contiguous pair -> ds_load_b64
                a[im] = *(const v2f*)&As[(wm + im * 16 + lm) * LDA + kk + kh * 2];
            #pragma unroll
            for (int in = 0; in < 4; ++in) { // strided pair -> ds_load_2addr
                const float* p = &Bs[(kk + kh * 2) * LDB + wn + in * 16 + lm];
                b[in][0] = p[0]; b[in][1] = p[LDB];
            }
            #pragma unroll
            for (int im = 0; im < 2; ++im)
                #pragma unroll
                for (int in = 0; in < 4; ++in)
                    c[im][in] = __builtin_amdgcn_wmma_f32_16x16x4_f32(
                        false, a[im], false, b[in], (short)0, c[im][in],
                        false, false);
        }
        __syncthreads();   // all reads done before this buffer is refilled
    }

    #pragma unroll
    for (int im = 0; im < 2; ++im)
        #pragma unroll
        for (int in = 0; in < 4; ++in) {
            int col  = wn + in * 16 + lm;
            int rowb = m0 + wm + im * 16 + kh * 8;
            #pragma unroll
            for (int i = 0; i < 8; ++i)
                dst[(size_t)(rowb + i) * K_ + col] = c[im][in][i];
        }
}

// ---------------------------------------------------------------------------
// Kernel 2: per-row gating.  s = routing[row]·W_g1 ; g = softmax(s*W_g2) ;
//           t[row,k] = xV[row,k] * sum_e g[e]*lambda_k[e,k]
// ---------------------------------------------------------------------------
__global__ __launch_bounds__(256)
void gate_kernel(const float* __restrict__ routing,
                 const float* __restrict__ xV,
                 const float* __restrict__ lambda_k,
                 const float* __restrict__ W_g1,
                 const float* __restrict__ W_g2,
                 float* __restrict__ t)
{
    const int row = blockIdx.x * 256 + threadIdx.x;
    const float* r = routing + (size_t)row * K_;
    float s = 0.0f;
    #pragma unroll 4
    for (int k = 0; k < K_; ++k) s += r[k] * W_g1[k];

    float g[E_];
    float mx = -INFINITY;
    #pragma unroll
    for (int e = 0; e < E_; ++e) { g[e] = s * W_g2[e]; mx = fmaxf(mx, g[e]); }
    float sum = 0.0f;
    #pragma unroll
    for (int e = 0; e < E_; ++e) { g[e] = __expf(g[e] - mx); sum += g[e]; }
    const float inv = 1.0f / sum;
    #pragma unroll
    for (int e = 0; e < E_; ++e) g[e] *= inv;

    const float* xv = xV + (size_t)row * K_;
    float* tr = t + (size_t)row * K_;
    #pragma unroll 4
    for (int k = 0; k < K_; ++k) {
        float lm = 0.0f;
        #pragma unroll
        for (int e = 0; e < E_; ++e) lm += g[e] * lambda_k[e * K_ + k];
        tr[k] = xv[k] * lm;
    }
}

// ---------------------------------------------------------------------------
// Kernel 3: out[n,d] = (sum_c x[n,c]*W[d,c] + sum_k t[n,k]*U_k[d,k]) * (1+v[d])
// One fused contraction of length 2048+128; W / U_k rows are the output dim,
// so the LDS B tile is out-major and fragment reads give the transpose.
// Double-buffered LDS + async copies; non-temporal output stores.
// ---------------------------------------------------------------------------
__global__ __launch_bounds__(256, 2)
void out_gemm_kernel(const float* __restrict__ x,
                     const float* __restrict__ W,
                     const float* __restrict__ t,
                     const float* __restrict__ U_k,
                     const float* __restrict__ v,
                     float* __restrict__ out)
{
    extern __shared__ float smem[];
    float* AsBase = smem;                    // 2 x [128*LDA]
    float* BsBase = smem + 2 * 128 * LDA;    // 2 x [128*LDA]

    const int tid  = threadIdx.x;
    const int lane = tid & 31;
    const int wid  = tid >> 5;
    const int kh   = lane >> 4;
    const int lm   = lane & 15;
    const int wm   = (wid >> 1) * 32;
    const int wn   = (wid & 1) * 64;
    const int m0   = blockIdx.y * 128;  // rows of x / out
    const int n0   = blockIdx.x * 128;  // cols of out = rows of W/U_k

    v8f c[2][4];
    #pragma unroll
    for (int im = 0; im < 2; ++im)
        #pragma unroll
        for (int in = 0; in < 4; ++in)
            #pragma unroll
            for (int q = 0; q < 8; ++q) c[im][in][q] = 0.0f;

    const int CHX = D_ / 32;          // 64 chunks over x/W
    const int CHT = K_ / 32;          // 4 chunks over t/U_k
    const int CH  = CHX + CHT;

    auto issue = [&](int ch, int buf) {
        float* As = AsBase + buf * (128 * LDA);
        float* Bs = BsBase + buf * (128 * LDA);
        if (ch < CHX) {
            const int c0 = ch * 32;
            #pragma unroll
            for (int i = 0; i < 4; ++i) {
                int idx = tid + i * 256;
                int r = idx >> 3, c4 = (idx & 7) * 4;
                cp_async16(x + (size_t)(m0 + r) * D_ + c0 + c4, &As[r * LDA + c4]);
                cp_async16(W + (size_t)(n0 + r) * D_ + c0 + c4, &Bs[r * LDA + c4]);
            }
        } else {
            const int c0 = (ch - CHX) * 32;
            #pragma unroll
            for (int i = 0; i < 4; ++i) {
                int idx = tid + i * 256;
                int r = idx >> 3, c4 = (idx & 7) * 4;
                cp_async16(t   + (size_t)(m0 + r) * K_ + c0 + c4, &As[r * LDA + c4]);
                cp_async16(U_k + (size_t)(n0 + r) * K_ + c0 + c4, &Bs[r * LDA + c4]);
            }
        }
    };

    issue(0, 0);
    for (int ch = 0; ch < CH; ++ch) {
        if (ch + 1 < CH) { issue(ch + 1, (ch + 1) & 1); cp_wait(1); }
        else             { cp_wait(0); }
        __syncthreads();

        const float* As = AsBase + (ch & 1) * (128 * LDA);
        const float* Bs = BsBase + (ch & 1) * (128 * LDA);
        #pragma unroll
        for (int kk = 0; kk < 32; kk += 4) {
            v2f a[2], b[4];
            #pragma unroll
            for (int im = 0; im < 2; ++im)   // contiguous pair -> ds_load_b64
                a[im] = *(const v2f*)&As[(wm + im * 16 + lm) * LDA + kk + kh * 2];
            #pragma unroll
            for (int in = 0; in < 4; ++in)   // contiguous pair -> ds_load_b64
                b[in] = *(const v2f*)&Bs[(wn + in * 16 + lm) * LDA + kk + kh * 2];
            #pragma unroll
            for (int im = 0; im < 2; ++im)
                #pragma unroll
                for (int in = 0; in < 4; ++in)
                    c[im][in] = __builtin_amdgcn_wmma_f32_16x16x4_f32(
                        false, a[im], false, b[in], (short)0, c[im][in],
                        false, false);
        }
        __syncthreads();
    }

    // ---- epilogue: scale by (1 + v[d]); stream out with non-temporal stores ----
    #pragma unroll
    for (int in = 0; in < 4; ++in) {
        const int col = n0 + wn + in * 16 + lm;
        const float sc = 1.0f + v[col];
        #pragma unroll
        for (int im = 0; im < 2; ++im) {
            const int rowb = m0 + wm + im * 16 + kh * 8;
            #pragma unroll
            for (int i = 0; i < 8; ++i)
                __builtin_nontemporal_store(c[im][in][i] * sc,
                                            out + (size_t)(rowb + i) * D_ + col);
        }
    }
}

// ---------------------------------------------------------------------------
extern "C" void kernel_launch(void* const* d_in, const int* in_sizes, int n_in,
                              void* d_out, int out_size, void* d_ws, size_t ws_size,
                              hipStream_t stream)
{
    const float* x        = (const float*)d_in[0];
    const float* W        = (const float*)d_in[1];
    const float* U_k      = (const float*)d_in[2];
    const float* V_k      = (const float*)d_in[3];
    const float* lambda_k = (const float*)d_in[4];
    const float* v        = (const float*)d_in[5];
    const float* W_g1     = (const float*)d_in[6];
    const float* W_g2     = (const float*)d_in[7];
    float* out = (float*)d_out;

    float* routing = (float*)d_ws;                       // [N, K]
    float* xV      = routing + (size_t)N_ * K_;          // [N, K]
    float* t       = xV      + (size_t)N_ * K_;          // [N, K]

    const size_t smem_xu  = (size_t)(2 * 128 * LDA + 2 * 32 * LDB) * sizeof(float);
    const size_t smem_out = (size_t)(4 * 128 * LDA) * sizeof(float);

    xu_gemm_kernel<<<dim3(N_ / 128), 256, smem_xu, stream>>>(x, U_k, routing);
    xu_gemm_kernel<<<dim3(N_ / 128), 256, smem_xu, stream>>>(x, V_k, xV);
    gate_kernel<<<dim3(N_ / 256), 256, 0, stream>>>(routing, xV, lambda_k,
                                                    W_g1, W_g2, t);
    out_gemm_kernel<<<dim3(D_ / 128, N_ / 128), 256, smem_out, stream>>>(
        x, W, t, U_k, v, out);
}